// BrightnessLoss_33552284517112
// MI455X (gfx1250) — compile-verified
//
#include <hip/hip_runtime.h>

typedef float v2f __attribute__((ext_vector_type(2)));
typedef float v4f __attribute__((ext_vector_type(4)));
typedef float v8f __attribute__((ext_vector_type(8)));

#define HW        262144      // 512*512 pixels per channel plane
#define IMG_STR   (3 * HW)    // floats per image (3 channels)
#define BLOCKS    2048
#define THREADS   256
#define NWAVES    (THREADS / 32)

__device__ __forceinline__ float clamp01(float x) {
    return __builtin_fminf(__builtin_fmaxf(x, 0.0f), 1.0f);
}

// |clamp01(max(pr,pg,pb)) - clamp01(max(tr,tg,tb))|
// (clamp is monotone, so clamp(max) == max(clamp) — saves 4 ops/pixel)
__device__ __forceinline__ float vdiff(float pr, float pg, float pb,
                                       float tr, float tg, float tb) {
    float vp = clamp01(__builtin_fmaxf(__builtin_fmaxf(pr, pg), pb));
    float vt = clamp01(__builtin_fmaxf(__builtin_fmaxf(tr, tg), tb));
    return __builtin_fabsf(vp - vt);
}

// Exact fp32 wave32 sum via V_WMMA_F32_16X16X4_F32.
// A: VGPR0 = per-lane acc, VGPR1 = 0  ->  A[m][0]=acc[m], A[m][2]=acc[m+16]
// B: all ones 4x16                    ->  D[m][*] = acc[m] + acc[m+16]
// c[r] lanes0-15 = rowsum(r), lanes16-31 = rowsum(r+8); sum 8 VGPRs, then
// fold halves with shfl_xor(16). EXEC must be all ones at the call site.
__device__ __forceinline__ float wave_reduce_wmma(float acc) {
    v2f a; a.x = acc;  a.y = 0.0f;
    v2f b; b.x = 1.0f; b.y = 1.0f;
    v8f c = {};
    c = __builtin_amdgcn_wmma_f32_16x16x4_f32(
            /*neg_a=*/false, a, /*neg_b=*/false, b,
            /*c_mod=*/(short)0, c, /*reuse_a=*/false, /*reuse_b=*/false);
    float t = ((c[0] + c[1]) + (c[2] + c[3])) + ((c[4] + c[5]) + (c[6] + c[7]));
    t += __shfl_xor(t, 16, 32);
    return t;
}

__global__ void __launch_bounds__(THREADS)
brightness_l1_partial(const float* __restrict__ pred,
                      const float* __restrict__ tgt,
                      float* __restrict__ partial,
                      int ngroups /* total_pixels / 4 */) {
    float acc = 0.0f;
    int tid    = blockIdx.x * blockDim.x + threadIdx.x;
    int stride = gridDim.x * blockDim.x;

    for (int g = tid; g < ngroups; g += stride) {
        int p = g << 2;                 // pixel index (HW % 4 == 0, never straddles images)
        int n = p / HW;
        int i = p - n * HW;
        size_t base = (size_t)n * IMG_STR + (size_t)i;

        v4f pr = __builtin_nontemporal_load((const v4f*)(pred + base));
        v4f pg = __builtin_nontemporal_load((const v4f*)(pred + base + HW));
        v4f pb = __builtin_nontemporal_load((const v4f*)(pred + base + 2 * HW));
        v4f tr = __builtin_nontemporal_load((const v4f*)(tgt  + base));
        v4f tg = __builtin_nontemporal_load((const v4f*)(tgt  + base + HW));
        v4f tb = __builtin_nontemporal_load((const v4f*)(tgt  + base + 2 * HW));

        acc += vdiff(pr.x, pg.x, pb.x, tr.x, tg.x, tb.x);
        acc += vdiff(pr.y, pg.y, pb.y, tr.y, tg.y, tb.y);
        acc += vdiff(pr.z, pg.z, pb.z, tr.z, tg.z, tb.z);
        acc += vdiff(pr.w, pg.w, pb.w, tr.w, tg.w, tb.w);
    }

    // wave32 reduction on the matrix pipe (all lanes active here)
    float wsum = wave_reduce_wmma(acc);

    __shared__ float lds[NWAVES];
    int lane = threadIdx.x & 31;
    int wave = threadIdx.x >> 5;
    if (lane == 0) lds[wave] = wsum;
    __syncthreads();
    if (threadIdx.x == 0) {
        float s = 0.0f;
        #pragma unroll
        for (int w = 0; w < NWAVES; ++w) s += lds[w];
        partial[blockIdx.x] = s;
    }
}

__global__ void __launch_bounds__(THREADS)
brightness_l1_final(const float* __restrict__ partial, int nparts,
                    float* __restrict__ out, float inv_count) {
    float acc = 0.0f;
    // nparts is a multiple of THREADS -> uniform trip count, EXEC stays full
    for (int i = threadIdx.x; i < nparts; i += THREADS) acc += partial[i];

    float wsum = wave_reduce_wmma(acc);

    __shared__ float lds[NWAVES];
    int lane = threadIdx.x & 31;
    int wave = threadIdx.x >> 5;
    if (lane == 0) lds[wave] = wsum;
    __syncthreads();
    if (threadIdx.x == 0) {
        float s = 0.0f;
        #pragma unroll
        for (int w = 0; w < NWAVES; ++w) s += lds[w];
        out[0] = s * inv_count;
    }
}

extern "C" void kernel_launch(void* const* d_in, const int* in_sizes, int n_in,
                              void* d_out, int out_size, void* d_ws, size_t ws_size,
                              hipStream_t stream) {
    const float* pred = (const float*)d_in[0];
    const float* tgt  = (const float*)d_in[1];
    float* out        = (float*)d_out;
    float* partial    = (float*)d_ws;          // BLOCKS floats of scratch

    long long nelem  = in_sizes[0];            // 32*3*512*512
    long long npix   = nelem / 3;              // 8,388,608
    int ngroups      = (int)(npix / 4);        // float4 groups
    float inv_count  = 1.0f / (float)npix;

    brightness_l1_partial<<<BLOCKS, THREADS, 0, stream>>>(pred, tgt, partial, ngroups);
    brightness_l1_final<<<1, THREADS, 0, stream>>>(partial, BLOCKS, out, inv_count);
}